// GCNLayer_57037165691114
// MI455X (gfx1250) — compile-verified
//
#include <hip/hip_runtime.h>
#include <hip/hip_bf16.h>

typedef __attribute__((ext_vector_type(2))) float v2f;
typedef __attribute__((ext_vector_type(8))) float v8f;

#define IN_DIM 64

// ---------------------------------------------------------------------------
// Kernel 1: zero the aggregation scratch buffer (float4 stores, full BW).
// ---------------------------------------------------------------------------
__global__ void gcn_zero_kernel(float4* __restrict__ p, int n4) {
    int i = blockIdx.x * blockDim.x + threadIdx.x;
    if (i < n4) p[i] = make_float4(0.f, 0.f, 0.f, 0.f);
}

// ---------------------------------------------------------------------------
// Kernel 2: edge scatter-add. 16 threads per edge; each thread moves 4 floats
// via one b128 load + 4 hardware f32 atomics (agent scope, relaxed order ->
// global_atomic_add_f32 resolved in L2).
// ---------------------------------------------------------------------------
__global__ void gcn_scatter_kernel(const float* __restrict__ feat,
                                   const int* __restrict__ src,
                                   const int* __restrict__ dst,
                                   float* __restrict__ agg,
                                   int n_edges) {
    int gid = blockIdx.x * blockDim.x + threadIdx.x;
    int e = gid >> 4;                 // 16 threads per edge
    if (e >= n_edges) return;
    int f = (gid & 15) * 4;           // 4 features per thread

    int s = src[e];
    int d = dst[e];

    const float4 v = *reinterpret_cast<const float4*>(feat + (long)s * IN_DIM + f);
    float* base = agg + (long)d * IN_DIM + f;

    __hip_atomic_fetch_add(base + 0, v.x, __ATOMIC_RELAXED, __HIP_MEMORY_SCOPE_AGENT);
    __hip_atomic_fetch_add(base + 1, v.y, __ATOMIC_RELAXED, __HIP_MEMORY_SCOPE_AGENT);
    __hip_atomic_fetch_add(base + 2, v.z, __ATOMIC_RELAXED, __HIP_MEMORY_SCOPE_AGENT);
    __hip_atomic_fetch_add(base + 3, v.w, __ATOMIC_RELAXED, __HIP_MEMORY_SCOPE_AGENT);
}

// ---------------------------------------------------------------------------
// Kernel 3: out = relu(agg @ W^T + b) using V_WMMA_F32_16X16X4_F32.
// One wave computes a 16-row strip of agg against all 64 output columns:
// 4 accumulator tiles (v8f each), K=64 done as 16 WMMA steps of K=4.
//
// A (16x4 f32) lane layout per ISA: lanes 0-15 hold M=lane, VGPR0=K0,VGPR1=K1;
// lanes 16-31 hold M=lane-16, VGPR0=K2,VGPR1=K3  -> contiguous float2 load.
// B (4x16) mirrors: VGPR0 = row K0 (lanes0-15)/K2 (lanes16-31), VGPR1 = K1/K3,
// N = lane&15. Since h[m,n] = sum_k agg[m,k]*W[n,k], B(k,n) = W[n*64 + k],
// which is also a contiguous float2 load from W's row n.
// C/D: VGPR r -> M = r + 8*(lane>=16), N = lane&15.
// ---------------------------------------------------------------------------
__global__ void gcn_wmma_gemm_kernel(const float* __restrict__ agg,
                                     const float* __restrict__ W,
                                     const float* __restrict__ bias,
                                     float* __restrict__ out,
                                     int n_rows) {
    const int lane = threadIdx.x & 31;
    const int wave = threadIdx.x >> 5;
    const int mtile = blockIdx.x * (blockDim.x >> 5) + wave;
    const int m0 = mtile * 16;
    if (m0 >= n_rows) return;         // wave-uniform: EXEC stays all-ones

    const int half = lane >> 4;       // 0: lanes 0-15, 1: lanes 16-31
    const int l16  = lane & 15;

    v8f acc[4] = {};                  // 4 N-tiles of 16 columns each

    const float* arow = agg + (long)(m0 + l16) * IN_DIM;
    const float* wrow0 = W + (long)(0 * 16 + l16) * IN_DIM;
    const float* wrow1 = W + (long)(1 * 16 + l16) * IN_DIM;
    const float* wrow2 = W + (long)(2 * 16 + l16) * IN_DIM;
    const float* wrow3 = W + (long)(3 * 16 + l16) * IN_DIM;

#pragma unroll
    for (int kb = 0; kb < 16; ++kb) {
        const int k = kb * 4 + half * 2;
        v2f a = *reinterpret_cast<const v2f*>(arow + k);

        v2f b0 = *reinterpret_cast<const v2f*>(wrow0 + k);
        v2f b1 = *reinterpret_cast<const v2f*>(wrow1 + k);
        v2f b2 = *reinterpret_cast<const v2f*>(wrow2 + k);
        v2f b3 = *reinterpret_cast<const v2f*>(wrow3 + k);

        acc[0] = __builtin_amdgcn_wmma_f32_16x16x4_f32(false, a, false, b0,
                                                       (short)0, acc[0], false, false);
        acc[1] = __builtin_amdgcn_wmma_f32_16x16x4_f32(false, a, false, b1,
                                                       (short)0, acc[1], false, false);
        acc[2] = __builtin_amdgcn_wmma_f32_16x16x4_f32(false, a, false, b2,
                                                       (short)0, acc[2], false, false);
        acc[3] = __builtin_amdgcn_wmma_f32_16x16x4_f32(false, a, false, b3,
                                                       (short)0, acc[3], false, false);
    }

    // Epilogue: bias + ReLU + store. Lanes 0-15 cover rows m0+r, lanes 16-31
    // rows m0+8+r; each half-wave stores 64 contiguous bytes per (t, r).
#pragma unroll
    for (int t = 0; t < 4; ++t) {
        const int n = t * 16 + l16;
        const float bv = bias[n];
#pragma unroll
        for (int r = 0; r < 8; ++r) {
            const int m = m0 + r + half * 8;
            float v = acc[t][r] + bv;
            out[(long)m * IN_DIM + n] = v > 0.f ? v : 0.f;
        }
    }
}

// ---------------------------------------------------------------------------
// Launcher
// ---------------------------------------------------------------------------
extern "C" void kernel_launch(void* const* d_in, const int* in_sizes, int n_in,
                              void* d_out, int out_size, void* d_ws, size_t ws_size,
                              hipStream_t stream) {
    const float* features = (const float*)d_in[0];   // [N, 64] f32
    const int*   src      = (const int*)d_in[1];     // [E] i32
    const int*   dst      = (const int*)d_in[2];     // [E] i32
    const float* W        = (const float*)d_in[3];   // [64, 64] f32
    const float* bias     = (const float*)d_in[4];   // [64] f32
    float*       out      = (float*)d_out;           // [N, 64] f32

    const int n_nodes = in_sizes[0] / IN_DIM;        // 50000
    const int n_edges = in_sizes[1];                 // 800000

    float* agg = (float*)d_ws;                       // 12.8 MB scratch

    // 1) agg = 0
    const int n4 = n_nodes * IN_DIM / 4;
    gcn_zero_kernel<<<(n4 + 255) / 256, 256, 0, stream>>>((float4*)agg, n4);

    // 2) agg[dst] += features[src]
    const long scat_threads = (long)n_edges * 16;
    gcn_scatter_kernel<<<(int)((scat_threads + 255) / 256), 256, 0, stream>>>(
        features, src, dst, agg, n_edges);

    // 3) out = relu(agg @ W^T + b)
    const int mtiles = (n_nodes + 15) / 16;          // 3125
    const int waves_per_block = 256 / 32;            // 8
    gcn_wmma_gemm_kernel<<<(mtiles + waves_per_block - 1) / waves_per_block,
                           256, 0, stream>>>(agg, W, bias, out, n_nodes);
}